// HyenaOperator_23476291240521
// MI455X (gfx1250) — compile-verified
//
#include <hip/hip_runtime.h>

#define FEATURES 768
#define ORDER 2
#define BATCH 2
#define SEQ 8192
#define IW (FEATURES * (ORDER + 2))   // 3072
#define MROWS (BATCH * SEQ)           // 16384
#define NFFT (2 * SEQ)                // 16384

typedef __attribute__((ext_vector_type(2))) float v2f;
typedef __attribute__((ext_vector_type(8))) float v8f;

// ---------------------------------------------------------------------------
// GEMM 1: up[M=16384, N=3072] = u[M,768] @ w_in[768,3072] + b_in
// One wave per 16x16 output tile, V_WMMA_F32_16X16X4_F32, K-loop step 4.
// Memory-bound at 23.3 TB/s, so fp32 matrix WMMA (no downconvert) is free.
// ---------------------------------------------------------------------------
__global__ void __launch_bounds__(256) gemm_in_kernel(
    const float* __restrict__ u, const float* __restrict__ w_in,
    const float* __restrict__ b_in, float* __restrict__ up) {
  const int lane = threadIdx.x & 31;
  const int wave = threadIdx.x >> 5;
  const int tile = blockIdx.x * 8 + wave;
  const int TN = IW / 16;                 // 192 tiles along N
  const int tm = (tile / TN) * 16;
  const int tn = (tile % TN) * 16;
  const int mrow = tm + (lane & 15);      // A: lanes 0-15/16-31 -> M = lane&15
  const int ncol = tn + (lane & 15);      // B: N = lane&15
  const int kq = (lane >> 4) << 1;        // K pair: 0-1 (lo half) / 2-3 (hi half)

  const float* arow = u + (size_t)mrow * FEATURES;
  v8f acc = {0.f, 0.f, 0.f, 0.f, 0.f, 0.f, 0.f, 0.f};
#pragma unroll 8
  for (int k = 0; k < FEATURES; k += 4) {
    if ((k & 127) == 0) {
      __builtin_prefetch(arow + k + 128, 0, 3);       // near-scope prefetch
      __builtin_prefetch(&w_in[(size_t)(k + 128) * IW + ncol], 0, 3);
    }
    v2f a, b;
    a.x = arow[k + kq];
    a.y = arow[k + kq + 1];
    b.x = w_in[(size_t)(k + kq) * IW + ncol];
    b.y = w_in[(size_t)(k + kq + 1) * IW + ncol];
    acc = __builtin_amdgcn_wmma_f32_16x16x4_f32(false, a, false, b,
                                                (short)0, acc, false, false);
  }
  // C/D layout: VGPR r, lanes 0-15 -> M=tm+r ; lanes 16-31 -> M=tm+8+r
  const int n = tn + (lane & 15);
  const int mbase = tm + ((lane >> 4) << 3);
  const float bb = b_in[n];
#pragma unroll
  for (int r = 0; r < 8; ++r) {
    up[(size_t)(mbase + r) * IW + n] = acc[r] + bb;
  }
}

// ---------------------------------------------------------------------------
// Depthwise causal conv (kernel 3, pad (2,0)) + split + v = v*x0.
// Writes v, x1, x2 in feature-major (b,f,t) layout for contiguous FFT rows.
// ---------------------------------------------------------------------------
__global__ void __launch_bounds__(256) dwconv_kernel(
    const float* __restrict__ up, const float* __restrict__ conv_k,
    const float* __restrict__ conv_b, float* __restrict__ vT,
    float* __restrict__ x1T, float* __restrict__ x2T) {
  int idx = blockIdx.x * blockDim.x + threadIdx.x;
  if (idx >= BATCH * SEQ * FEATURES) return;
  int f = idx % FEATURES;
  int t = (idx / FEATURES) % SEQ;
  int b = idx / (FEATURES * SEQ);
  float uc[4];
#pragma unroll
  for (int c = 0; c < 4; ++c) {
    int ch = c * FEATURES + f;
    size_t base = (size_t)(b * SEQ + t) * IW + ch;
    float acc = conv_k[2 * IW + ch] * up[base];
    if (t >= 1) acc += conv_k[1 * IW + ch] * up[base - IW];
    if (t >= 2) acc += conv_k[0 * IW + ch] * up[base - 2 * (size_t)IW];
    uc[c] = acc + conv_b[ch];
  }
  size_t o = (size_t)(b * FEATURES + f) * SEQ + t;
  x1T[o] = uc[1];
  x2T[o] = uc[2];
  vT[o] = uc[3] * uc[0];    // v * x0
}

// Transpose h (ORDER,1,L,F) -> (ORDER,F,L) for contiguous per-feature reads.
__global__ void __launch_bounds__(256) htrans_kernel(
    const float* __restrict__ h, float* __restrict__ hT) {
  int idx = blockIdx.x * blockDim.x + threadIdx.x;
  if (idx >= ORDER * SEQ * FEATURES) return;
  int f = idx % FEATURES;
  int t = (idx / FEATURES) % SEQ;
  int o = idx / (FEATURES * SEQ);
  hT[(size_t)(o * FEATURES + f) * SEQ + t] =
      h[(size_t)(o * SEQ + t) * FEATURES + f];
}

// ---------------------------------------------------------------------------
// Per-(batch,feature) 2-order fftconv chain entirely in LDS (320 KB/WGP).
// Forward DIF (natural -> bit-reversed), elementwise product in permuted
// order, inverse DIT (bit-reversed -> natural). Net: linear causal conv.
// While order 0 computes its FFTs, order 1's filter row is prefetched into
// LDS with GLOBAL_LOAD_ASYNC_TO_LDS_B128 (ASYNCcnt), hiding its HBM latency.
// ---------------------------------------------------------------------------
__global__ void __launch_bounds__(256) fftconv_kernel(
    float* __restrict__ vT, const float* __restrict__ hT,
    const float* __restrict__ bias, const float* __restrict__ x1T,
    const float* __restrict__ x2T) {
  extern __shared__ float lds[];
  float* vr = lds;                       // NFFT
  float* vi = lds + NFFT;                // NFFT
  float* hr = lds + 2 * NFFT;            // NFFT
  float* hi = lds + 3 * NFFT;            // NFFT
  float* vo = lds + 4 * NFFT;            // SEQ  (time-domain v for bias skip)
  float* hp = lds + 4 * NFFT + SEQ;      // SEQ  (async-prefetched next h)

  const int f = blockIdx.x;
  const int b = blockIdx.y;
  const int tid = threadIdx.x;
  const float inv_n = 1.0f / (float)NFFT;
  const size_t vrow = (size_t)(b * FEATURES + f) * SEQ;

  for (int t = tid; t < SEQ; t += 256) {
    float x = vT[vrow + t];
    vr[t] = x; vi[t] = 0.f; vo[t] = x;
    vr[t + SEQ] = 0.f; vi[t + SEQ] = 0.f;
  }

  for (int o = 0; o < ORDER; ++o) {
    if (o == 0) {
      // Synchronous load of h[0]; queue async LDS prefetch of h[1] so the
      // Tensor/async path overlaps with the 14 forward-FFT stages below.
      const size_t hrow = (size_t)(0 * FEATURES + f) * SEQ;
      for (int t = tid; t < SEQ; t += 256) {
        hr[t] = hT[hrow + t]; hi[t] = 0.f;
        hr[t + SEQ] = 0.f; hi[t + SEQ] = 0.f;
      }
      const float* hsrc = hT + (size_t)(1 * FEATURES + f) * SEQ;
      for (int t0 = tid * 4; t0 < SEQ; t0 += 1024) {
        unsigned la = (unsigned)(unsigned long long)(hp + t0);   // LDS offset
        unsigned long long ga = (unsigned long long)(hsrc + t0); // global addr
        asm volatile("global_load_async_to_lds_b128 %0, %1, off"
                     :: "v"(la), "v"(ga) : "memory");
      }
    } else {
      asm volatile("s_wait_asynccnt 0x0" ::: "memory");
      __syncthreads();
      for (int t = tid; t < SEQ; t += 256) {
        hr[t] = hp[t]; hi[t] = 0.f;
        hr[t + SEQ] = 0.f; hi[t + SEQ] = 0.f;
      }
    }
    __syncthreads();

    // Forward DIF FFT on v and h simultaneously (shared twiddles).
    for (int len = NFFT; len >= 2; len >>= 1) {
      const int half = len >> 1;
      const float base = -6.28318530717958647692f / (float)len;
      for (int idx = tid; idx < NFFT / 2; idx += 256) {
        const int blk = idx / half;
        const int j = idx - blk * half;
        const int i0 = blk * len + j;
        const int i1 = i0 + half;
        float s, c;
        __sincosf(base * (float)j, &s, &c);
        {
          float ar = vr[i0], ai = vi[i0];
          float br = vr[i1], bi = vi[i1];
          vr[i0] = ar + br; vi[i0] = ai + bi;
          float tr = ar - br, ti = ai - bi;
          vr[i1] = tr * c - ti * s;
          vi[i1] = tr * s + ti * c;
        }
        {
          float ar = hr[i0], ai = hi[i0];
          float br = hr[i1], bi = hi[i1];
          hr[i0] = ar + br; hi[i0] = ai + bi;
          float tr = ar - br, ti = ai - bi;
          hr[i1] = tr * c - ti * s;
          hi[i1] = tr * s + ti * c;
        }
      }
      __syncthreads();
    }

    // Pointwise product (both arrays identically bit-reversed).
    for (int t = tid; t < NFFT; t += 256) {
      float ar = vr[t], ai = vi[t];
      float br = hr[t], bi = hi[t];
      vr[t] = ar * br - ai * bi;
      vi[t] = ar * bi + ai * br;
    }
    __syncthreads();

    // Inverse DIT FFT (bit-reversed input -> natural output).
    for (int len = 2; len <= NFFT; len <<= 1) {
      const int half = len >> 1;
      const float base = 6.28318530717958647692f / (float)len;
      for (int idx = tid; idx < NFFT / 2; idx += 256) {
        const int blk = idx / half;
        const int j = idx - blk * half;
        const int i0 = blk * len + j;
        const int i1 = i0 + half;
        float s, c;
        __sincosf(base * (float)j, &s, &c);
        float br = vr[i1] * c - vi[i1] * s;
        float bi = vr[i1] * s + vi[i1] * c;
        float ar = vr[i0], ai = vi[i0];
        vr[i0] = ar + br; vi[i0] = ai + bi;
        vr[i1] = ar - br; vi[i1] = ai - bi;
      }
      __syncthreads();
    }

    // y = conv/N + v*bias ; v_new = y * x[o+1]
    const float bo = bias[o * FEATURES + f];
    const float* xg = (o == 0) ? x1T : x2T;
    for (int t = tid; t < SEQ; t += 256) {
      float y = vr[t] * inv_n + vo[t] * bo;
      vo[t] = y * xg[vrow + t];
    }
    __syncthreads();

    if (o + 1 < ORDER) {
      for (int t = tid; t < SEQ; t += 256) {
        vr[t] = vo[t]; vi[t] = 0.f;
        vr[t + SEQ] = 0.f; vi[t + SEQ] = 0.f;
      }
    } else {
      for (int t = tid; t < SEQ; t += 256) vT[vrow + t] = vo[t];
    }
  }
}

// ---------------------------------------------------------------------------
// GEMM 2: out[M=16384, N=768] = v[M,768] @ w_out[768,768] + b_out
// A is read from the feature-major vT layout (A[m][k] = vT[b,k,t]).
// ---------------------------------------------------------------------------
__global__ void __launch_bounds__(256) gemm_out_kernel(
    const float* __restrict__ vT, const float* __restrict__ w_out,
    const float* __restrict__ b_out, float* __restrict__ out) {
  const int lane = threadIdx.x & 31;
  const int wave = threadIdx.x >> 5;
  const int tile = blockIdx.x * 8 + wave;
  const int TN = FEATURES / 16;           // 48
  const int tm = (tile / TN) * 16;
  const int tn = (tile % TN) * 16;
  const int mrow = tm + (lane & 15);
  const int b = mrow >> 13;               // / SEQ
  const int t = mrow & (SEQ - 1);
  const int ncol = tn + (lane & 15);
  const int kq = (lane >> 4) << 1;

  v8f acc = {0.f, 0.f, 0.f, 0.f, 0.f, 0.f, 0.f, 0.f};
#pragma unroll 8
  for (int k = 0; k < FEATURES; k += 4) {
    if ((k & 127) == 0) {
      __builtin_prefetch(&w_out[(size_t)(k + 128) * FEATURES + ncol], 0, 3);
    }
    v2f a, bm;
    a.x = vT[(size_t)(b * FEATURES + k + kq) * SEQ + t];
    a.y = vT[(size_t)(b * FEATURES + k + kq + 1) * SEQ + t];
    bm.x = w_out[(size_t)(k + kq) * FEATURES + ncol];
    bm.y = w_out[(size_t)(k + kq + 1) * FEATURES + ncol];
    acc = __builtin_amdgcn_wmma_f32_16x16x4_f32(false, a, false, bm,
                                                (short)0, acc, false, false);
  }
  const int n = tn + (lane & 15);
  const int mbase = tm + ((lane >> 4) << 3);
  const float bb = b_out[n];
#pragma unroll
  for (int r = 0; r < 8; ++r) {
    out[(size_t)(mbase + r) * FEATURES + n] = acc[r] + bb;
  }
}

// ---------------------------------------------------------------------------
extern "C" void kernel_launch(void* const* d_in, const int* in_sizes, int n_in,
                              void* d_out, int out_size, void* d_ws,
                              size_t ws_size, hipStream_t stream) {
  (void)in_sizes; (void)n_in; (void)out_size; (void)ws_size;
  const float* u      = (const float*)d_in[0];
  const float* h      = (const float*)d_in[1];
  const float* bias   = (const float*)d_in[2];
  const float* w_in   = (const float*)d_in[3];
  const float* b_in   = (const float*)d_in[4];
  const float* conv_k = (const float*)d_in[5];
  const float* conv_b = (const float*)d_in[6];
  const float* w_out  = (const float*)d_in[7];
  const float* b_out  = (const float*)d_in[8];
  float* out = (float*)d_out;

  // Workspace carve-up (floats): up | vT | x1T | x2T | hT  (~403 MB total)
  float* ws = (float*)d_ws;
  const size_t row = (size_t)BATCH * FEATURES * SEQ;   // 12.58M floats
  float* up  = ws;                                     // MROWS*IW
  float* vT  = up + (size_t)MROWS * IW;
  float* x1T = vT + row;
  float* x2T = x1T + row;
  float* hT  = x2T + row;

  {
    int tiles = (MROWS / 16) * (IW / 16);              // 196608
    gemm_in_kernel<<<tiles / 8, 256, 0, stream>>>(u, w_in, b_in, up);
  }
  {
    int n = BATCH * SEQ * FEATURES;
    dwconv_kernel<<<(n + 255) / 256, 256, 0, stream>>>(up, conv_k, conv_b, vT,
                                                       x1T, x2T);
  }
  {
    int n = ORDER * SEQ * FEATURES;
    htrans_kernel<<<(n + 255) / 256, 256, 0, stream>>>(h, hT);
  }
  {
    dim3 grid(FEATURES, BATCH);
    size_t shmem = (size_t)(4 * NFFT + 2 * SEQ) * sizeof(float);   // 320 KB
    fftconv_kernel<<<grid, 256, shmem, stream>>>(vT, hT, bias, x1T, x2T);
  }
  {
    int tiles = (MROWS / 16) * (FEATURES / 16);        // 49152
    gemm_out_kernel<<<tiles / 8, 256, 0, stream>>>(vT, w_out, b_out, out);
  }
}